// NeuralNet_79542794322095
// MI455X (gfx1250) — compile-verified
//
#include <hip/hip_runtime.h>
#include <hip/hip_bf16.h>

typedef __attribute__((ext_vector_type(2))) float v2f;
typedef __attribute__((ext_vector_type(8))) float v8f;

#define GATES 400      // 4 * HIDDEN
#define HIDDEN 100
#define INDIM 13

// ---------------------------------------------------------------------------
// Kernels 1/3: xg = X @ W^T + (b0 + b1)   using V_WMMA_F32_16X16X4_F32
//   X:  [T, DIN] row-major;  W: [400, DIN] row-major (B = W^T)
//   XG: [T+pad, 400]
// One wave computes one 16(t) x 16(gate) tile; K accumulated in chunks of 4.
// A 16x4 f32 layout : lane l -> m = l&15, k = 4*ch + (l>>4)*2 + {0,1}
// B 4x16 f32 layout : lane l -> n = l&15, k = 4*ch + (l>>4)*2 + {0,1}
// C 16x16 f32 layout: vgpr j, lane l -> row = (l>>4)*8 + j, col = l&15
// DIN is a compile-time constant -> fully unrolled, branchless inner loop:
//   full chunks (DIN/4) need no bounds checks; odd tail uses clamped loads.
// ---------------------------------------------------------------------------
template <int DIN>
__global__ void gates_gemm_kernel(const float* __restrict__ X,
                                  const float* __restrict__ W,
                                  const float* __restrict__ b0,
                                  const float* __restrict__ b1,
                                  float* __restrict__ XG, int T)
{
    const int lane  = threadIdx.x & 31;
    const int wave  = threadIdx.x >> 5;
    const int mtile = blockIdx.x * (blockDim.x >> 5) + wave;
    const int t0    = mtile * 16;
    if (t0 >= T) return;                       // wave-uniform, EXEC stays full
    const int n0 = blockIdx.y * 16;

    const int m  = lane & 15;                  // A row / B col within tile
    const int kb = (lane >> 4) * 2;            // k sub-offset for this lane half

    const float* __restrict__ xrow = X + (size_t)(t0 + m) * DIN;
    const float* __restrict__ wrow = W + (size_t)(n0 + m) * DIN;

    constexpr int FULLCH = DIN / 4;            // fully in-bounds chunks
    v8f acc = {};

#pragma unroll
    for (int ch = 0; ch < FULLCH; ++ch) {
        const int k0 = ch * 4;
        v2f a, b;
        if constexpr ((DIN % 2) == 0) {
            // 8B-aligned (DIN even): one b64 load per operand.
            a = *(const v2f*)(xrow + k0 + kb);
            b = *(const v2f*)(wrow + k0 + kb);
        } else {
            const int ka = k0 + kb;            // in-bounds: ka+1 <= 4*FULLCH-1 < DIN
            a.x = xrow[ka];  a.y = xrow[ka + 1];
            b.x = wrow[ka];  b.y = wrow[ka + 1];
        }
        acc = __builtin_amdgcn_wmma_f32_16x16x4_f32(
                  false, a, false, b, (short)0, acc, false, false);
    }

    if constexpr ((DIN % 4) != 0) {
        // Tail chunk: clamped, branchless scalar loads + v_cndmask zeroing.
        const int k0  = FULLCH * 4;
        const int ka  = k0 + kb;
        const int ka1 = ka + 1;
        const int ca  = (ka  < DIN) ? ka  : (DIN - 1);
        const int ca1 = (ka1 < DIN) ? ka1 : (DIN - 1);
        const float ax = xrow[ca], ay = xrow[ca1];
        const float bx = wrow[ca], by = wrow[ca1];
        v2f a, b;
        a.x = (ka  < DIN) ? ax : 0.0f;
        a.y = (ka1 < DIN) ? ay : 0.0f;
        b.x = (ka  < DIN) ? bx : 0.0f;
        b.y = (ka1 < DIN) ? by : 0.0f;
        acc = __builtin_amdgcn_wmma_f32_16x16x4_f32(
                  false, a, false, b, (short)0, acc, false, false);
    }

    const int col    = n0 + m;
    const float bias = b0[col] + b1[col];
    const int row0   = t0 + ((lane >> 4) * 8);
#pragma unroll
    for (int j = 0; j < 8; ++j) {
        XG[(size_t)(row0 + j) * GATES + col] = acc[j] + bias;
    }
}

// ---------------------------------------------------------------------------
// Kernels 2/4: sequential LSTM scan, one workgroup (latency-bound).
//   - W_hh row j register-resident in thread j (100 VGPRs, no per-step W
//     traffic; h is the only LDS read, broadcast & conflict-free).
//   - Per-gate nonlinearity applied in the WIDE (400-thread) phase via the
//     identity tanh(x) = 2*sigmoid(2x) - 1  ->  act = alpha*sig(beta*x)+gamma,
//     branchless, one v_exp per gate across 13 waves in parallel.
//   - Narrow phase (100 threads): c = af*c + ai*ag; h = ao*tanh(c)  (one
//     transcendental on the critical path).
//   - xg(t+1) prefetched unconditionally (XG padded), 2 barriers/step.
// ---------------------------------------------------------------------------
__device__ __forceinline__ float tanh_f(float x) { return 1.0f - 2.0f / (__expf(2.0f * x) + 1.0f); }

__global__ void __launch_bounds__(512, 1)
lstm_scan_kernel(const float* __restrict__ XG,    // [T+1(+pad), 400]
                 const float* __restrict__ Whh,   // [400, 100]
                 float* __restrict__ Hout,        // [T, 100]
                 int T)
{
    __shared__ float h_lds[HIDDEN + 4];
    __shared__ float gbuf[GATES];

    const int j = threadIdx.x;

    // Gate j in [200,300) is the 'g' gate -> tanh; others -> sigmoid.
    const bool  is_g  = (j >= 200) && (j < 300);
    const float beta  = is_g ? 2.0f : 1.0f;     // also alpha
    const float gamma = is_g ? -1.0f : 0.0f;

    // Register-resident recurrent weights: row j of W_hh.
    float w[HIDDEN];
    if (j < GATES) {
#pragma unroll
        for (int k = 0; k < HIDDEN; ++k) w[k] = Whh[(size_t)j * HIDDEN + k];
    }
    if (j < HIDDEN + 4) h_lds[j] = 0.0f;
    float c_state = 0.0f;
    float xg_cur = XG[j];                        // step-0 pre-activation (padded)
    __syncthreads();

    for (int t = 0; t < T; ++t) {
        // Unconditional prefetch of next step (XG has a padded extra row).
        const float xg_next = XG[(size_t)(t + 1) * GATES + j];

        if (j < GATES) {
            float a0 = 0.f, a1 = 0.f, a2 = 0.f, a3 = 0.f;
#pragma unroll
            for (int k = 0; k < HIDDEN; k += 4) {
                a0 = fmaf(w[k    ], h_lds[k    ], a0);
                a1 = fmaf(w[k + 1], h_lds[k + 1], a1);
                a2 = fmaf(w[k + 2], h_lds[k + 2], a2);
                a3 = fmaf(w[k + 3], h_lds[k + 3], a3);
            }
            const float g = xg_cur + ((a0 + a1) + (a2 + a3));
            // act = sigmoid(g) for i,f,o ; tanh(g) = 2*sigmoid(2g)-1 for g-gate
            const float s = 1.0f / (1.0f + __expf(-beta * g));
            gbuf[j] = fmaf(beta, s, gamma);      // beta==alpha
        }
        __syncthreads();

        if (j < HIDDEN) {
            const float ai = gbuf[j];
            const float af = gbuf[j + 100];
            const float ag = gbuf[j + 200];
            const float ao = gbuf[j + 300];
            c_state = fmaf(af, c_state, ai * ag);
            const float hn = ao * tanh_f(c_state);
            h_lds[j] = hn;
            Hout[(size_t)t * HIDDEN + j] = hn;   // fire-and-forget store
        }
        __syncthreads();

        xg_cur = xg_next;
    }
}

// ---------------------------------------------------------------------------
// Kernel 5: out[t] = dot(h2[t], W_fc) + b_fc ; one wave per timestep.
// ---------------------------------------------------------------------------
__global__ void fc_kernel(const float* __restrict__ H,     // [T, 100]
                          const float* __restrict__ Wfc,   // [100]
                          const float* __restrict__ bfc,   // [1]
                          float* __restrict__ out, int T)
{
    const int lane = threadIdx.x & 31;
    const int wave = threadIdx.x >> 5;
    const int t = blockIdx.x * (blockDim.x >> 5) + wave;
    if (t >= T) return;

    float s = 0.0f;
#pragma unroll
    for (int n0 = 0; n0 < 128; n0 += 32) {
        const int n = n0 + lane;
        if (n < HIDDEN) s = fmaf(H[(size_t)t * HIDDEN + n], Wfc[n], s);
    }
#pragma unroll
    for (int off = 16; off > 0; off >>= 1) s += __shfl_down(s, off, 32);
    if (lane == 0) out[t] = s + bfc[0];
}

// ---------------------------------------------------------------------------
// Host-side launch
// ---------------------------------------------------------------------------
extern "C" void kernel_launch(void* const* d_in, const int* in_sizes, int n_in,
                              void* d_out, int out_size, void* d_ws, size_t ws_size,
                              hipStream_t stream)
{
    const float* x     = (const float*)d_in[0];   // [T, 13]
    const float* W_ih1 = (const float*)d_in[1];   // [400, 13]
    const float* W_hh1 = (const float*)d_in[2];   // [400, 100]
    const float* b_ih1 = (const float*)d_in[3];   // [400]
    const float* b_hh1 = (const float*)d_in[4];   // [400]
    const float* W_ih2 = (const float*)d_in[5];   // [400, 100]
    const float* W_hh2 = (const float*)d_in[6];   // [400, 100]
    const float* b_ih2 = (const float*)d_in[7];   // [400]
    const float* b_hh2 = (const float*)d_in[8];   // [400]
    const float* W_fc  = (const float*)d_in[9];   // [1, 100]
    const float* b_fc  = (const float*)d_in[10];  // [1]

    const int T = in_sizes[0] / INDIM;            // 131072

    // Workspace (f32): xg [T+1 rows + 512 pad] (reused), h1 [T,100], h2 [T,100]
    float* xg = (float*)d_ws;
    float* h1 = xg + (size_t)(T + 1) * GATES + 512;
    float* h2 = h1 + (size_t)T * HIDDEN;

    dim3 gblk(128);                                // 4 waves / block
    dim3 ggrd((unsigned)(T / 64), GATES / 16);     // wave per 16x16 tile

    // Layer 1: xg1 = x @ W_ih1^T + b ; scan -> h1
    gates_gemm_kernel<INDIM><<<ggrd, gblk, 0, stream>>>(x, W_ih1, b_ih1, b_hh1, xg, T);
    lstm_scan_kernel<<<1, 512, 0, stream>>>(xg, W_hh1, h1, T);

    // Layer 2: xg2 = h1 @ W_ih2^T + b ; scan -> h2
    gates_gemm_kernel<HIDDEN><<<ggrd, gblk, 0, stream>>>(h1, W_ih2, b_ih2, b_hh2, xg, T);
    lstm_scan_kernel<<<1, 512, 0, stream>>>(xg, W_hh2, h2, T);

    // Output projection
    fc_kernel<<<dim3((unsigned)((T + 7) / 8)), 256, 0, stream>>>(h2, W_fc, b_fc,
                                                                 (float*)d_out, T);
}